// Net_2491081031714
// MI455X (gfx1250) — compile-verified
//
#include <hip/hip_runtime.h>
#include <hip/hip_bf16.h>
#include <math.h>

typedef __attribute__((ext_vector_type(2))) float v2f;
typedef __attribute__((ext_vector_type(8))) float v8f;

#define BATCH   16384
#define N_IN    784
#define N_TH    9      // only first 9 thetas survive to the output
#define OUT_DIM 10

// One wave (32 lanes) reduces 16 rows via V_WMMA_F32_16X16X4_F32 with B = ones:
// D[m][n] accumulates sum_k A[m][k] = sum of squares of row m. 196 k-steps of 4.
__global__ __launch_bounds__(256) void polar_net_kernel(
    const float* __restrict__ x,
    const float* __restrict__ scale1, const float* __restrict__ rot1,
    const float* __restrict__ scale2, const float* __restrict__ rot2,
    const float* __restrict__ scale3, const float* __restrict__ rot3,
    float* __restrict__ out)
{
    const int lane  = threadIdx.x & 31;
    const int wave  = (blockIdx.x * blockDim.x + threadIdx.x) >> 5;
    const int row0  = wave * 16;
    if (row0 >= BATCH) return;              // wave-uniform: EXEC stays all-ones

    // A-matrix 16x4 f32 layout: lanes 0-15 -> M=lane, K={0,1}; lanes 16-31 -> M=lane-16, K={2,3}
    const int   myrow = row0 + (lane & 15);
    const float* p    = x + (size_t)myrow * N_IN + ((lane >> 4) << 1);  // col offset 0 or 2

    v2f bones; bones.x = 1.0f; bones.y = 1.0f;   // B = ones(4x16): layout-independent
    v8f acc0 = {}, acc1 = {}, acc2 = {}, acc3 = {};

    // 784 cols = 196 k-blocks of 4; 4 accumulators break the WMMA RAW chain
    for (int c = 0; c < 196; c += 4) {
        v2f a0 = *(const v2f*)(p + 4 * (c + 0));
        v2f a1 = *(const v2f*)(p + 4 * (c + 1));
        v2f a2 = *(const v2f*)(p + 4 * (c + 2));
        v2f a3 = *(const v2f*)(p + 4 * (c + 3));
        a0 *= a0; a1 *= a1; a2 *= a2; a3 *= a3;  // square elementwise -> sum-of-squares
        acc0 = __builtin_amdgcn_wmma_f32_16x16x4_f32(false, a0, false, bones, (short)0, acc0, false, false);
        acc1 = __builtin_amdgcn_wmma_f32_16x16x4_f32(false, a1, false, bones, (short)0, acc1, false, false);
        acc2 = __builtin_amdgcn_wmma_f32_16x16x4_f32(false, a2, false, bones, (short)0, acc2, false, false);
        acc3 = __builtin_amdgcn_wmma_f32_16x16x4_f32(false, a3, false, bones, (short)0, acc3, false, false);
    }
    v8f acc = acc0 + acc1 + acc2 + acc3;

    // D layout: VGPR v holds M=v (lanes 0-15) and M=v+8 (lanes 16-31); all N columns equal.
    // Give lane m (0..15) the row-sum for row row0+m. Done with full EXEC.
    float S = 0.0f;
    #pragma unroll
    for (int v = 0; v < 8; ++v) {
        float lo = __shfl(acc[v], 0, 32);    // row-sum of M=v
        float hi = __shfl(acc[v], 16, 32);   // row-sum of M=v+8
        if (lane == v)     S = lo;
        if (lane == v + 8) S = hi;
    }

    if (lane < 16) {
        const int    row = row0 + lane;
        const float* xr  = x + (size_t)row * N_IN;

        float xv[N_TH];
        #pragma unroll
        for (int k = 0; k < N_TH; ++k) xv[k] = xr[k];   // hot in L2 from the stream above

        // r chain
        float r = sqrtf(fmaxf(S, 0.0f));
        r = fmaxf(r * scale1[0], 0.0f);
        r = fmaxf(r * scale2[0], 0.0f);
        r = r * scale3[0];

        // theta chain: th_k = relu(relu(acos(clip(x_k / sqrt(s_k))) + rot1_k) + rot2_k) + rot3_k
        float th[N_TH];
        float pref = 0.0f;                    // prefix sum of squares of x_0..x_{k-1}
        #pragma unroll
        for (int k = 0; k < N_TH; ++k) {
            float sk    = fmaxf(S - pref, 0.0f);          // suffix sum-of-squares
            float denom = sqrtf(sk);
            float ratio = xv[k] / denom;
            ratio = fminf(1.0f, fmaxf(-1.0f, ratio));
            float t = acosf(ratio);
            t = fmaxf(t + rot1[k], 0.0f);
            t = fmaxf(t + rot2[k], 0.0f);
            th[k] = t + rot3[k];
            pref += xv[k] * xv[k];
        }

        // polar_to_cartesian: pp[m] = prod_{j<m} sin(th_j)
        float sn[N_TH], cs[N_TH];
        #pragma unroll
        for (int k = 0; k < N_TH; ++k) { sn[k] = sinf(th[k]); cs[k] = cosf(th[k]); }

        float pp[N_TH + 1];
        pp[0] = 1.0f;
        #pragma unroll
        for (int m = 1; m <= N_TH; ++m) pp[m] = pp[m - 1] * sn[m - 1];

        float cart[OUT_DIM];
        cart[0] = r * cs[8] * pp[8];
        cart[1] = r * sn[8] * pp[8];
        #pragma unroll
        for (int j = 0; j < 8; ++j) cart[2 + j] = r * cs[7 - j] * pp[7 - j];

        // softmax over 10
        float mx = cart[0];
        #pragma unroll
        for (int i = 1; i < OUT_DIM; ++i) mx = fmaxf(mx, cart[i]);
        float e[OUT_DIM], sum = 0.0f;
        #pragma unroll
        for (int i = 0; i < OUT_DIM; ++i) { e[i] = expf(cart[i] - mx); sum += e[i]; }
        float inv = 1.0f / sum;

        float* op = out + (size_t)row * OUT_DIM;
        #pragma unroll
        for (int i = 0; i < OUT_DIM; ++i) op[i] = e[i] * inv;
    }
}

extern "C" void kernel_launch(void* const* d_in, const int* in_sizes, int n_in,
                              void* d_out, int out_size, void* d_ws, size_t ws_size,
                              hipStream_t stream) {
    const float* x      = (const float*)d_in[0];
    const float* scale1 = (const float*)d_in[1];
    const float* rot1   = (const float*)d_in[2];
    const float* scale2 = (const float*)d_in[3];
    const float* rot2   = (const float*)d_in[4];
    const float* scale3 = (const float*)d_in[5];
    const float* rot3   = (const float*)d_in[6];
    float* out = (float*)d_out;

    const int waves   = BATCH / 16;                 // 1024 waves, 16 rows each
    const int threads = 256;                        // 8 wave32 per block
    const int blocks  = (waves * 32 + threads - 1) / threads;  // 128
    polar_net_kernel<<<blocks, threads, 0, stream>>>(x, scale1, rot1, scale2, rot2,
                                                     scale3, rot3, out);
}